// PlainGCN_90941637525586
// MI455X (gfx1250) — compile-verified
//
#include <hip/hip_runtime.h>
#include <hip/hip_bf16.h>

#define NNODES 100000
#define NEDGES 1600000
#define FDIM   128
#define FOUT   40
#define FOUTP  48
#define BN_EPS 1e-5f

typedef __attribute__((ext_vector_type(2))) float v2f;
typedef __attribute__((ext_vector_type(8))) float v8f;

// ---------------------------------------------------------------- utilities
__global__ void fill_f32(float* __restrict__ p, float v, int n) {
    int i = blockIdx.x * blockDim.x + threadIdx.x;
    if (i < n) p[i] = v;
}

__global__ void deg_count(const int* __restrict__ dst, float* __restrict__ deg, int e) {
    int i = blockIdx.x * blockDim.x + threadIdx.x;
    if (i < e) atomicAdd(&deg[dst[i]], 1.0f);
}

__global__ void make_dinv(float* __restrict__ deg, int n) {
    int i = blockIdx.x * blockDim.x + threadIdx.x;
    if (i < n) {
        float d = deg[i];
        deg[i] = (d > 0.0f) ? rsqrtf(d) : 0.0f;
    }
}

__global__ void pad_w3(const float* __restrict__ W3, float* __restrict__ wp) {
    int i = blockIdx.x * blockDim.x + threadIdx.x;
    if (i < FDIM * FOUTP) {
        int k = i / FOUTP, c = i % FOUTP;
        wp[i] = (c < FOUT) ? W3[k * FOUT + c] : 0.0f;
    }
}

// ---------------------------------------------------------------- WMMA GEMM
// C[nrows, NT*16] = X[nrows, 128] @ W[128, NT*16], f32 WMMA 16x16x4.
// One block = 256 threads = 8 wave32; each wave owns a 16-row tile and all
// NT column tiles. W is staged in LDS K-pair-interleaved so every B fragment
// is one aligned ds_load_b64; A fragments are unconditional global b64 loads
// from a clamped row (out-of-range rows only feed D rows that are never
// stored, so no masking is needed).
template <int NT>
__global__ __launch_bounds__(256)
void gemm_wmma(const float* __restrict__ X, const float* __restrict__ W,
               float* __restrict__ C, int nrows) {
    constexpr int NC = NT * 16;
    // layout: lw[(k>>1)*(2*NC) + c*2 + (k&1)]
    __shared__ float lw[FDIM * NC];

    for (int i = threadIdx.x; i < FDIM * NC; i += 256) {
        int k = i / NC, c = i - k * NC;
        lw[(k >> 1) * (2 * NC) + c * 2 + (k & 1)] = W[i];
    }
    __syncthreads();

    const int lane = threadIdx.x & 31;
    const int wave = threadIdx.x >> 5;
    const int half = lane >> 4;     // 0: K+0/K+1 lanes, 1: K+2/K+3 lanes
    const int r    = lane & 15;     // row-in-tile for A, col-in-tile for B/C
    const int rowbase = blockIdx.x * 128 + wave * 16;

    v8f acc[NT];
    const v8f vzero = {0.f, 0.f, 0.f, 0.f, 0.f, 0.f, 0.f, 0.f};
#pragma unroll
    for (int t = 0; t < NT; ++t) acc[t] = vzero;

    int arow = rowbase + r;
    int asrc = arow < nrows ? arow : (nrows - 1);   // clamp instead of mask
    const float* __restrict__ xrow = X + (long long)asrc * FDIM;

#pragma unroll 4
    for (int k0 = 0; k0 < FDIM; k0 += 4) {
        const int kk = k0 + half * 2;
        v2f a = *(const v2f*)(xrow + kk);
        const float* __restrict__ lrow = lw + (kk >> 1) * (2 * NC);
#pragma unroll
        for (int t = 0; t < NT; ++t) {
            v2f b = *(const v2f*)(lrow + (t * 16 + r) * 2);
            acc[t] = __builtin_amdgcn_wmma_f32_16x16x4_f32(
                false, a, false, b, (short)0, acc[t], false, false);
        }
    }

    // C/D layout: VGPR vr -> M = vr (lanes 0-15) / vr+8 (lanes 16-31), N = lane&15
#pragma unroll
    for (int t = 0; t < NT; ++t) {
#pragma unroll
        for (int vr = 0; vr < 8; ++vr) {
            int row = rowbase + vr + half * 8;
            if (row < nrows) C[(long long)row * NC + t * 16 + r] = acc[t][vr];
        }
    }
}

// ------------------------------------------------------- graph aggregation
// agg[n] = h[n] * dinv[n]^2  (self-loop term; also initializes the buffer)
__global__ void agg_selfinit(const float4* __restrict__ h4, const float* __restrict__ dinv,
                             float4* __restrict__ agg4, int ncols, int n) {
    long long i = (long long)blockIdx.x * blockDim.x + threadIdx.x;
    long long total4 = (long long)n * ncols / 4;
    if (i < total4) {
        int node = (int)((i * 4) / ncols);
        float di = dinv[node];
        float s = di * di;
        float4 v = h4[i];
        v.x *= s; v.y *= s; v.z *= s; v.w *= s;
        agg4[i] = v;
    }
}

// one wave per edge: agg[dst] += h[src] * dinv[src]*dinv[dst]
__global__ __launch_bounds__(256)
void agg_edges(const int* __restrict__ src, const int* __restrict__ dst,
               const float* __restrict__ dinv, const float* __restrict__ h,
               float* __restrict__ agg, int ncols, int e) {
    int gw   = (int)(((long long)blockIdx.x * blockDim.x + threadIdx.x) >> 5);
    int lane = threadIdx.x & 31;
    if (gw >= e) return;
    int s = src[gw], d = dst[gw];
    float norm = dinv[s] * dinv[d];
    const float4* __restrict__ hs = (const float4*)(h + (long long)s * ncols);
    float* __restrict__ ad = agg + (long long)d * ncols;
    const int nq = ncols >> 2;
    for (int f = lane; f < nq; f += 32) {
        float4 v = hs[f];
        atomicAdd(&ad[f * 4 + 0], v.x * norm);
        atomicAdd(&ad[f * 4 + 1], v.y * norm);
        atomicAdd(&ad[f * 4 + 2], v.z * norm);
        atomicAdd(&ad[f * 4 + 3], v.w * norm);
    }
}

// ---------------------------------------------------------------- batchnorm
#define BN_ROWS 500
__global__ __launch_bounds__(256)
void bn_stats(const float* __restrict__ x, float* __restrict__ stats, int n) {
    int f   = threadIdx.x & 127;
    int sub = threadIdx.x >> 7;       // 0..1
    int rbeg = blockIdx.x * BN_ROWS;
    int rend = rbeg + BN_ROWS; if (rend > n) rend = n;
    float s = 0.0f, ss = 0.0f;
    for (int r = rbeg + sub; r < rend; r += 2) {
        float v = x[(long long)r * FDIM + f];
        s += v; ss += v * v;
    }
    __shared__ float ls[256], lss[256];
    ls[threadIdx.x] = s; lss[threadIdx.x] = ss;
    __syncthreads();
    if (sub == 0) {
        s  += ls[threadIdx.x + 128];
        ss += lss[threadIdx.x + 128];
        atomicAdd(&stats[f], s);
        atomicAdd(&stats[128 + f], ss);
    }
}

__global__ void bn_finalize(float* __restrict__ stats, int n) {
    int f = threadIdx.x;              // 128 threads
    float inv_n = 1.0f / (float)n;
    float mu  = stats[f] * inv_n;
    float var = stats[128 + f] * inv_n - mu * mu;
    stats[256 + f] = mu;
    stats[384 + f] = rsqrtf(var + BN_EPS);
}

__global__ void bn_apply_relu(const float* __restrict__ agg, const float* __restrict__ stats,
                              const float* __restrict__ g, const float* __restrict__ be,
                              float* __restrict__ out, int n) {
    long long i = (long long)blockIdx.x * blockDim.x + threadIdx.x;
    if (i < (long long)n * FDIM) {
        int f = (int)(i & (FDIM - 1));
        float v = (agg[i] - stats[256 + f]) * stats[384 + f] * g[f] + be[f];
        out[i] = v > 0.0f ? v : 0.0f;
    }
}

// ----------------------------------------------------------------- softmax
__global__ void softmax40(const float* __restrict__ agg3, const float* __restrict__ b3,
                          float* __restrict__ out, int n) {
    int i = blockIdx.x * blockDim.x + threadIdx.x;
    if (i >= n) return;
    const float* __restrict__ a = agg3 + (long long)i * FOUTP;
    float l[FOUT];
    float m = -3.0e38f;
#pragma unroll
    for (int c = 0; c < FOUT; ++c) { l[c] = a[c] + b3[c]; m = fmaxf(m, l[c]); }
    float s = 0.0f;
#pragma unroll
    for (int c = 0; c < FOUT; ++c) { l[c] = __expf(l[c] - m); s += l[c]; }
    float inv = 1.0f / s;
#pragma unroll
    for (int c = 0; c < FOUT; ++c) out[(long long)i * FOUT + c] = l[c] * inv;
}

// ------------------------------------------------------------------ launch
extern "C" void kernel_launch(void* const* d_in, const int* in_sizes, int n_in,
                              void* d_out, int out_size, void* d_ws, size_t ws_size,
                              hipStream_t stream) {
    const float* x   = (const float*)d_in[0];
    const int*   ei  = (const int*)d_in[1];
    const int*   src = ei;
    const int*   dst = ei + NEDGES;
    const float* W1  = (const float*)d_in[2];
    const float* g1  = (const float*)d_in[4];
    const float* be1 = (const float*)d_in[5];
    const float* W2  = (const float*)d_in[6];
    const float* g2  = (const float*)d_in[8];
    const float* be2 = (const float*)d_in[9];
    const float* W3  = (const float*)d_in[10];
    const float* b3  = (const float*)d_in[11];
    float* out = (float*)d_out;

    float* ws    = (float*)d_ws;
    float* dinv  = ws;
    float* buf1  = ws + NNODES;
    float* buf2  = buf1 + (size_t)NNODES * FDIM;
    float* stats = buf2 + (size_t)NNODES * FDIM;
    float* w3p   = stats + 512;

    const int T = 256;
    const int gN    = (NNODES + T - 1) / T;
    const int gE    = (NEDGES + T - 1) / T;
    const int gNF   = (int)(((long long)NNODES * FDIM + T - 1) / T);
    const int gNF4  = (int)(((long long)NNODES * FDIM / 4 + T - 1) / T);
    const int gNF34 = (int)(((long long)NNODES * FOUTP / 4 + T - 1) / T);
    const int gEw   = (int)(((long long)NEDGES * 32 + T - 1) / T);
    const int gGemm = (NNODES + 127) / 128;

    // gcn_norm: deg (with self loop) -> dinv = rsqrt(deg)
    fill_f32<<<gN, T, 0, stream>>>(dinv, 1.0f, NNODES);
    deg_count<<<gE, T, 0, stream>>>(dst, dinv, NEDGES);
    make_dinv<<<gN, T, 0, stream>>>(dinv, NNODES);
    pad_w3<<<(FDIM * FOUTP + T - 1) / T, T, 0, stream>>>(W3, w3p);

    // ---- layer 1:  h1=x@W1 -> aggregate -> BN+ReLU  (b1 cancels inside BN)
    gemm_wmma<8><<<gGemm, T, 0, stream>>>(x, W1, buf1, NNODES);
    agg_selfinit<<<gNF4, T, 0, stream>>>((const float4*)buf1, dinv, (float4*)buf2, FDIM, NNODES);
    agg_edges<<<gEw, T, 0, stream>>>(src, dst, dinv, buf1, buf2, FDIM, NEDGES);
    fill_f32<<<1, T, 0, stream>>>(stats, 0.0f, 256);
    bn_stats<<<(NNODES + BN_ROWS - 1) / BN_ROWS, T, 0, stream>>>(buf2, stats, NNODES);
    bn_finalize<<<1, 128, 0, stream>>>(stats, NNODES);
    bn_apply_relu<<<gNF, T, 0, stream>>>(buf2, stats, g1, be1, buf1, NNODES);

    // ---- layer 2
    gemm_wmma<8><<<gGemm, T, 0, stream>>>(buf1, W2, buf2, NNODES);
    agg_selfinit<<<gNF4, T, 0, stream>>>((const float4*)buf2, dinv, (float4*)buf1, FDIM, NNODES);
    agg_edges<<<gEw, T, 0, stream>>>(src, dst, dinv, buf2, buf1, FDIM, NEDGES);
    fill_f32<<<1, T, 0, stream>>>(stats, 0.0f, 256);
    bn_stats<<<(NNODES + BN_ROWS - 1) / BN_ROWS, T, 0, stream>>>(buf1, stats, NNODES);
    bn_finalize<<<1, 128, 0, stream>>>(stats, NNODES);
    bn_apply_relu<<<gNF, T, 0, stream>>>(buf1, stats, g2, be2, buf2, NNODES);

    // ---- layer 3: h3 = x3 @ W3pad (128x48), aggregate, +b3, softmax
    float* h3   = buf1;                               // N x 48
    float* agg3 = buf1 + (size_t)NNODES * 64;         // N x 48 (disjoint)
    gemm_wmma<3><<<gGemm, T, 0, stream>>>(buf2, w3p, h3, NNODES);
    agg_selfinit<<<gNF34, T, 0, stream>>>((const float4*)h3, dinv, (float4*)agg3, FOUTP, NNODES);
    agg_edges<<<gEw, T, 0, stream>>>(src, dst, dinv, h3, agg3, FOUTP, NEDGES);
    softmax40<<<(NNODES + 127) / 128, 128, 0, stream>>>(agg3, b3, out, NNODES);
}